// CTCLayer_49263274885962
// MI455X (gfx1250) — compile-verified
//
#include <hip/hip_runtime.h>
#include <stdint.h>

// Problem constants (from the reference)
#define B_    512
#define T_    512
#define C_    128
#define L_    64
#define S_    129          // 2L+1 extended states
#define TU_   128          // input_len = C  (source bug replicated)
#define BLANK_ 127
#define NEG_  (-1e30f)
#define EPS_  (1e-7f)

// TDM staging: 16 rows of 128 floats per chunk, double buffered
#define ROWS_ 16
#define NCH_  (TU_ / ROWS_)   // 8 chunks
#define NTHREADS_ 160         // 5 wave32s; threads 0..128 own states

typedef unsigned int v4u __attribute__((ext_vector_type(4)));
typedef int          v8i __attribute__((ext_vector_type(8)));
typedef int          v4i __attribute__((ext_vector_type(4)));

// Issue one Tensor-Data-Mover DMA: copy a ROWS_ x C_ float tile (row-major,
// row stride C_) from global memory into LDS.  Descriptor per CDNA5 ISA
// §8.3-8.5.  Tracked by TENSORcnt; EXEC is ignored by the TDM op so this must
// only be reached by one wave (guarded by the caller).
__device__ __forceinline__ void tdm_load_rows(const float* gsrc, void* lds_dst) {
  uint64_t ga  = (uint64_t)(uintptr_t)gsrc;
  // low 32 bits of a flat LDS pointer are the LDS byte offset (ISA §10.2)
  uint32_t lds = (uint32_t)(uintptr_t)lds_dst;

  v4u g0;
  g0.x = 1u;                                                // count=1 (valid), user mode
  g0.y = lds;                                               // lds_addr
  g0.z = (uint32_t)ga;                                      // global_addr[31:0]
  g0.w = ((uint32_t)(ga >> 32) & 0x01FFFFFFu) | (2u << 30); // global_addr[56:32] | type=2

  v8i g1;
  g1[0] = (int)(2u << 16);                 // wg_mask=0, data_size=2 (4 bytes)
  g1[1] = (int)((uint32_t)C_ << 16);       // tensor_dim0[15:0]=128  (bits 63:48)
  g1[2] = (int)((uint32_t)ROWS_ << 16);    // tensor_dim0 hi=0 | tensor_dim1[15:0]=ROWS
  g1[3] = (int)((uint32_t)C_ << 16);       // tensor_dim1 hi=0 | tile_dim0=128
  g1[4] = (int)(ROWS_);                    // tile_dim1=ROWS | tile_dim2=0
  g1[5] = (int)(C_);                       // tensor_dim0_stride[31:0]=128 elements
  g1[6] = 0;                               // stride hi / dim1_stride lo
  g1[7] = 0;

  v4i z4 = {0, 0, 0, 0};                   // groups 2/3 unused (2D tensor)
#if __clang_major__ >= 23
  v8i z8 = {0, 0, 0, 0, 0, 0, 0, 0};
  __builtin_amdgcn_tensor_load_to_lds(g0, g1, z4, z4, z8, 0);
#else
  __builtin_amdgcn_tensor_load_to_lds(g0, g1, z4, z4, 0);
#endif
}

__global__ __launch_bounds__(NTHREADS_)
void ctc_loss_kernel(const int* __restrict__ y_true,
                     const float* __restrict__ y_pred,
                     float* __restrict__ out) {
  __shared__ float rowbuf[2][ROWS_][C_];   // 16 KB: double-buffered y_pred rows
  __shared__ float alpha[2][S_ + 3];       // ping-pong alpha
  __shared__ int   labs[L_];

  const int b   = blockIdx.x;
  const int tid = threadIdx.x;
  const int s   = tid;                     // extended-state index this thread owns
  const float* base = y_pred + (size_t)b * T_ * C_;

  // Stage labels for this batch element
  if (tid < L_) labs[tid] = y_true[b * L_ + tid];

  // Wave 0 kicks off the first TDM chunk (rows 0..ROWS_-1)
  if (tid < 32) {
    tdm_load_rows(base, &rowbuf[0][0][0]);
  }
  __syncthreads();                         // labs visible to all waves

  // Per-state constants: ext[s] and skip_ok[s]
  int  ext_s = BLANK_;
  bool skip  = false;
  if (s < S_ && (s & 1)) {
    int k = s >> 1;
    ext_s = labs[k];
    skip  = (s >= 3) && (ext_s != labs[k - 1]);
  }

  for (int c = 0; c < NCH_; ++c) {
    if (tid < 32) {
      // Software pipeline: launch the next DMA *before* waiting on the current
      // one (TDM ops are in-order per wave), so chunk c+1 streams in while we
      // compute on chunk c.  Destination buffer was last read in chunk c-1 and
      // is protected by the per-row barriers below.
      if (c + 1 < NCH_) {
        tdm_load_rows(base + (size_t)(c + 1) * ROWS_ * C_,
                      &rowbuf[(c + 1) & 1][0][0]);
        __builtin_amdgcn_s_wait_tensorcnt(1);  // chunk c landed; c+1 in flight
      } else {
        __builtin_amdgcn_s_wait_tensorcnt(0);  // final chunk landed
      }
    }
    __syncthreads();                           // release all waves on chunk c

    const float (*rows)[C_] = rowbuf[c & 1];
    for (int r = 0; r < ROWS_; ++r) {
      const int t = c * ROWS_ + r;
      if (s < S_) {
        // lp[t,s] = log(y_pred[b,t,ext[s]] + eps), gathered from the LDS row
        const float lp = logf(rows[r][ext_s] + EPS_);
        if (t == 0) {
          alpha[0][s] = (s <= 1) ? lp : NEG_;
        } else {
          const float* ap = alpha[(t - 1) & 1];
          const float a0 = ap[s];
          const float a1 = (s >= 1) ? ap[s - 1] : NEG_;
          const float a2 = skip    ? ap[s - 2] : NEG_;
          // lse3 with the reference's NEG floor
          const float m = fmaxf(fmaxf(fmaxf(a0, a1), a2), NEG_);
          const float v = m + logf(expf(a0 - m) + expf(a1 - m) + expf(a2 - m));
          alpha[t & 1][s] = v + lp;
        }
      }
      __syncthreads();
    }
  }

  // loss = -logsumexp(alphaT[S-2], alphaT[S-1]); final parity = (TU_-1)&1
  if (tid == 0) {
    const float* af = alpha[(TU_ - 1) & 1];
    const float x = af[S_ - 2];
    const float y = af[S_ - 1];
    const float m = fmaxf(x, y);
    out[b] = -(m + logf(expf(x - m) + expf(y - m)));
  }
}

extern "C" void kernel_launch(void* const* d_in, const int* in_sizes, int n_in,
                              void* d_out, int out_size, void* d_ws, size_t ws_size,
                              hipStream_t stream) {
  const int*   y_true = (const int*)d_in[0];    // [512, 64] int32
  const float* y_pred = (const float*)d_in[1];  // [512, 512, 128] float32
  float*       out    = (float*)d_out;          // [512, 1] float32

  ctc_loss_kernel<<<dim3(B_), dim3(NTHREADS_), 0, stream>>>(y_true, y_pred, out);
}